// CGCNNConv_3496103379076
// MI455X (gfx1250) — compile-verified
//
#include <hip/hip_runtime.h>
#include <math.h>

#define N_NODES 50000
#define N_EDGES 800000
#define NF 64
#define EF 32
#define M2 128          // 2*NF
#define EPS 1e-5f

typedef __attribute__((ext_vector_type(2))) float v2f;
typedef __attribute__((ext_vector_type(8))) float v8f;

__device__ __forceinline__ v8f wmma_f32_k4(v2f a, v2f b, v8f c) {
  // V_WMMA_F32_16X16X4_F32: D = A(16x4) * B(4x16) + C(16x16)
  return __builtin_amdgcn_wmma_f32_16x16x4_f32(
      /*neg_a=*/false, a, /*neg_b=*/false, b,
      /*c_mod=*/(short)0, c, /*reuse_a=*/false, /*reuse_b=*/false);
}

__device__ __forceinline__ float softplus_f(float x) {
  return x > 20.f ? x : log1pf(__expf(x));
}
__device__ __forceinline__ float sigmoid_f(float x) {
  return 1.f / (1.f + __expf(-x));
}

// ---------------------------------------------------------------------------
// zero scratch (node accumulator + BN stat accumulators)
// ---------------------------------------------------------------------------
__global__ void zero_kernel(float* __restrict__ p, int n) {
  int i = blockIdx.x * blockDim.x + threadIdx.x;
  if (i < n) p[i] = 0.f;
}

// ---------------------------------------------------------------------------
// h_src = nf @ W_src ; h_dst = nf @ W_dst   (biases absorbed by edge BN)
// 512 threads = 16 waves; waves 0-7 -> W_src N-tiles, waves 8-15 -> W_dst.
// Each block computes 16 nodes x 128 cols for both projections.
// ---------------------------------------------------------------------------
__global__ __launch_bounds__(512) void node_proj_kernel(
    const float* __restrict__ nf, const float* __restrict__ Wsrc,
    const float* __restrict__ Wdst, float* __restrict__ hsrc,
    float* __restrict__ hdst) {
  __shared__ float a_sh[16 * NF];            // 4 KB node tile
  const int tid = threadIdx.x;
  const int nbase = blockIdx.x * 16;
  {
    const v2f* g = (const v2f*)(nf + (size_t)nbase * NF);
    ((v2f*)a_sh)[tid] = g[tid];              // 1024 floats / 512 threads
  }
  __syncthreads();

  const int wave = tid >> 5;
  const int lane = tid & 31;
  const int half = lane >> 4;                // 0 or 1
  const int l16  = lane & 15;
  const int koff = half * 2;                 // K sub-offset for A/B fragments
  const int ntile = wave & 7;
  const int col = ntile * 16 + l16;
  const float* W = (wave < 8) ? Wsrc : Wdst;
  float* out     = (wave < 8) ? hsrc : hdst;

  v8f acc = {0.f, 0.f, 0.f, 0.f, 0.f, 0.f, 0.f, 0.f};
#pragma unroll
  for (int k0 = 0; k0 < NF; k0 += 4) {
    v2f a;
    a.x = a_sh[l16 * NF + k0 + koff];
    a.y = a_sh[l16 * NF + k0 + koff + 1];
    v2f b;
    b.x = W[(size_t)(k0 + koff) * M2 + col];
    b.y = W[(size_t)(k0 + koff + 1) * M2 + col];
    acc = wmma_f32_k4(a, b, acc);
  }
#pragma unroll
  for (int r = 0; r < 8; ++r)
    out[(size_t)(nbase + r + 8 * half) * M2 + col] = acc[r];
}

// ---------------------------------------------------------------------------
// Edge pipeline. APPLY=false: compute m tiles, accumulate per-channel
// sum/sumsq (edge BN stats). APPLY=true: recompute m (cheaper than storing
// 410MB; h_src/h_dst gathers hit the 192MB L2), apply BN + sigmoid*softplus
// gate, scatter-add to node accumulator.
// One wave owns a 16-edge tile x all 128 output channels (8 WMMA N-tiles).
// ---------------------------------------------------------------------------
template <bool APPLY>
__global__ __launch_bounds__(256) void edge_pass_kernel(
    const float* __restrict__ ef, const int* __restrict__ src,
    const int* __restrict__ dst, const float* __restrict__ We,
    const float* __restrict__ hsrc, const float* __restrict__ hdst,
    const float* __restrict__ gamma_m, const float* __restrict__ beta_m,
    float* __restrict__ e_stats, float* __restrict__ node_h) {
  __shared__ float w_sh[EF * M2];            // 16 KB: W_edge staged once
  __shared__ float sred[2 * M2];             // block BN-stat reduction
  const int tid = threadIdx.x;
  {
    const float4* g = (const float4*)We;     // 4096 floats, 16 per thread
    float4* s = (float4*)w_sh;
#pragma unroll
    for (int j = 0; j < 4; ++j) s[j * 256 + tid] = g[j * 256 + tid];
  }
  if (!APPLY) sred[tid] = 0.f;
  __syncthreads();

  const int lane = tid & 31;
  const int half = lane >> 4;
  const int l16  = lane & 15;
  const int koff = half * 2;
  const int gw = blockIdx.x * 8 + (tid >> 5);
  const int nw = gridDim.x * 8;
  const int TILES = N_EDGES / 16;

  float lsum[8], lssq[8], scl[8], sft[8];
  if (!APPLY) {
#pragma unroll
    for (int nt = 0; nt < 8; ++nt) { lsum[nt] = 0.f; lssq[nt] = 0.f; }
  } else {
    const float inv = 1.f / (float)N_EDGES;
#pragma unroll
    for (int nt = 0; nt < 8; ++nt) {         // per-lane cols fixed across loop
      const int c = nt * 16 + l16;
      float mean = e_stats[c] * inv;
      float var  = e_stats[M2 + c] * inv - mean * mean;
      float rstd = rsqrtf(var + EPS);
      scl[nt] = rstd * gamma_m[c];
      sft[nt] = beta_m[c] - mean * scl[nt];
    }
  }

  for (int t = gw; t < TILES; t += nw) {
    const int ebase = t * 16;
    if (t + nw < TILES)
      __builtin_prefetch(ef + (size_t)(ebase + nw * 16) * EF, 0, 1);

    v2f a[8];                                // full 16x32 A tile in regs
#pragma unroll
    for (int kk = 0; kk < 8; ++kk)
      a[kk] = *(const v2f*)(ef + (size_t)(ebase + l16) * EF + kk * 4 + koff);

    int se[8], de[8];                        // indices for this lane's C rows
#pragma unroll
    for (int r = 0; r < 8; ++r) {
      const int e = ebase + r + 8 * half;
      se[r] = src[e];
      de[r] = dst[e];
    }

    v8f acc[8];
#pragma unroll
    for (int nt = 0; nt < 8; ++nt) {
      v8f c = {0.f, 0.f, 0.f, 0.f, 0.f, 0.f, 0.f, 0.f};
      const int cN = nt * 16 + l16;
#pragma unroll
      for (int kk = 0; kk < 8; ++kk) {
        const int k = kk * 4 + koff;
        v2f b;
        b.x = w_sh[k * M2 + cN];
        b.y = w_sh[(k + 1) * M2 + cN];
        c = wmma_f32_k4(a[kk], b, c);
      }
      acc[nt] = c;
    }

    // m += h_src[src] + h_dst[dst]  (gathers; 51MB working set, L2-resident)
#pragma unroll
    for (int nt = 0; nt < 8; ++nt) {
      const int cN = nt * 16 + l16;
#pragma unroll
      for (int r = 0; r < 8; ++r)
        acc[nt][r] += hsrc[(size_t)se[r] * M2 + cN] +
                      hdst[(size_t)de[r] * M2 + cN];
    }

    if (!APPLY) {
#pragma unroll
      for (int nt = 0; nt < 8; ++nt)
#pragma unroll
        for (int r = 0; r < 8; ++r) {
          float v = acc[nt][r];
          lsum[nt] += v;
          lssq[nt] += v * v;
        }
    } else {
#pragma unroll
      for (int nt = 0; nt < 4; ++nt) {       // cols 0-63 gate cols 64-127
        const int c = nt * 16 + l16;
#pragma unroll
        for (int r = 0; r < 8; ++r) {
          float f = acc[nt][r] * scl[nt] + sft[nt];
          float s = acc[nt + 4][r] * scl[nt + 4] + sft[nt + 4];
          float o = sigmoid_f(f) * softplus_f(s);
          atomicAdd(&node_h[(size_t)de[r] * NF + c], o);   // segment_sum
        }
      }
    }
  }

  if (!APPLY) {
#pragma unroll
    for (int nt = 0; nt < 8; ++nt) {         // ds_add_f32 block reduction
      const int c = nt * 16 + l16;
      atomicAdd(&sred[c], lsum[nt]);
      atomicAdd(&sred[M2 + c], lssq[nt]);
    }
    __syncthreads();
    atomicAdd(&e_stats[tid], sred[tid]);     // 256 global atomics / block
  }
}

// ---------------------------------------------------------------------------
// Node BN stats: per-channel sum/sumsq of the 50000x64 accumulator.
// ---------------------------------------------------------------------------
__global__ __launch_bounds__(256) void node_stats_kernel(
    const float* __restrict__ node_h, float* __restrict__ n_stats) {
  __shared__ float sred[2 * NF];
  const int tid = threadIdx.x;
  if (tid < 2 * NF) sred[tid] = 0.f;
  __syncthreads();
  const int c = tid & (NF - 1);              // stride is multiple of 64
  float lsum = 0.f, lssq = 0.f;
  const int total = N_NODES * NF;
  for (int i = blockIdx.x * blockDim.x + tid; i < total;
       i += gridDim.x * blockDim.x) {
    float v = node_h[i];
    lsum += v;
    lssq += v * v;
  }
  atomicAdd(&sred[c], lsum);
  atomicAdd(&sred[NF + c], lssq);
  __syncthreads();
  if (tid < 2 * NF) atomicAdd(&n_stats[tid], sred[tid]);
}

// ---------------------------------------------------------------------------
// out = softplus(node_feats + BN(node_h))
// ---------------------------------------------------------------------------
__global__ void final_kernel(const float* __restrict__ nf,
                             const float* __restrict__ node_h,
                             const float* __restrict__ n_stats,
                             const float* __restrict__ gamma_n,
                             const float* __restrict__ beta_n,
                             float* __restrict__ out) {
  const int i = blockIdx.x * blockDim.x + threadIdx.x;
  if (i >= N_NODES * NF) return;
  const int c = i & (NF - 1);
  const float inv = 1.f / (float)N_NODES;
  float mean = n_stats[c] * inv;
  float var  = n_stats[NF + c] * inv - mean * mean;
  float rstd = rsqrtf(var + EPS);
  float h = (node_h[i] - mean) * rstd * gamma_n[c] + beta_n[c];
  out[i] = softplus_f(nf[i] + h);
}

// ---------------------------------------------------------------------------
extern "C" void kernel_launch(void* const* d_in, const int* in_sizes, int n_in,
                              void* d_out, int out_size, void* d_ws,
                              size_t ws_size, hipStream_t stream) {
  (void)in_sizes; (void)n_in; (void)out_size; (void)ws_size;
  const float* node_feats = (const float*)d_in[0];
  const float* edge_feats = (const float*)d_in[1];
  const int*   src        = (const int*)d_in[2];
  const int*   dst        = (const int*)d_in[3];
  const float* W_src      = (const float*)d_in[4];
  // d_in[5] b_src, d_in[7] b_dst, d_in[9] b_edge: per-column constants are
  // exactly cancelled by the edge BatchNorm mean subtraction -> unused.
  const float* W_dst      = (const float*)d_in[6];
  const float* W_edge     = (const float*)d_in[8];
  const float* gamma_m    = (const float*)d_in[10];
  const float* beta_m     = (const float*)d_in[11];
  const float* gamma_n    = (const float*)d_in[12];
  const float* beta_n     = (const float*)d_in[13];
  float* out = (float*)d_out;

  float* ws      = (float*)d_ws;
  float* h_src   = ws;                               // N*128 = 25.6 MB
  float* h_dst   = ws + (size_t)N_NODES * M2;        // N*128 = 25.6 MB
  float* node_h  = ws + (size_t)N_NODES * 2 * M2;    // N*64  = 12.8 MB
  float* e_stats = node_h + (size_t)N_NODES * NF;    // 256 floats
  float* n_stats = e_stats + 2 * M2;                 // 128 floats

  const int nz = N_NODES * NF + 2 * M2 + 2 * NF;     // accumulator + stats
  zero_kernel<<<(nz + 255) / 256, 256, 0, stream>>>(node_h, nz);

  node_proj_kernel<<<N_NODES / 16, 512, 0, stream>>>(
      node_feats, W_src, W_dst, h_src, h_dst);

  edge_pass_kernel<false><<<1024, 256, 0, stream>>>(
      edge_feats, src, dst, W_edge, h_src, h_dst, gamma_m, beta_m,
      e_stats, node_h);
  edge_pass_kernel<true><<<1024, 256, 0, stream>>>(
      edge_feats, src, dst, W_edge, h_src, h_dst, gamma_m, beta_m,
      e_stats, node_h);

  node_stats_kernel<<<512, 256, 0, stream>>>(node_h, n_stats);

  final_kernel<<<(N_NODES * NF + 255) / 256, 256, 0, stream>>>(
      node_feats, node_h, n_stats, gamma_n, beta_n, out);
}